// Code_Multiplexing_22522808500314
// MI455X (gfx1250) — compile-verified
//
#include <hip/hip_runtime.h>

// ---------------------------------------------------------------------------
// Hermite/Hadamard 4-point complex butterfly over a 524288x4 batch.
// Memory-bound: 64MB in + 64MB out => ~5.5us floor at 23.3 TB/s.
// Math: per (b,l), y = Z4 * x with Z4 entries in {+-1, +-i}. As reals this is
// an 8x8 {0,+-1} matrix M8 applied to u = (Re x0, Im x0, ..., Im x3).
// Implemented as two chained V_WMMA_F32_16X16X4_F32 ops (K=8 split into 2x4),
// exact in f32. Data rides in the A operand so plain coalesced b64 loads land
// in the required VGPR striping with zero cross-lane shuffles.
//
// Round-2 refinement: all load pointers / store offsets are hoisted out of the
// tile loop and advanced by wave-uniform strides (32-bit store offsets from a
// uniform base -> SADDR-form stores), removing the per-iteration 64-bit
// address arithmetic seen in round 1's disassembly.
// ---------------------------------------------------------------------------

typedef __attribute__((ext_vector_type(2))) float v2f;
typedef __attribute__((ext_vector_type(8))) float v8f;

// M8[n, j]  with n = 2q + c (complex out-row q, c=0 Re / 1 Im)
//           and  j = 2s + d (stream s, d=0 Re / 1 Im)
//   Z4[q,s] = h * (q odd ? i : 1),  h = (-1)^popcount(q & s)
//   q even: M8 = (c==d) ? h : 0
//   q odd : M8 = (c!=d) ? (c ? +h : -h) : 0      (i*(a+bi) = -b + ai)
__device__ __forceinline__ float m8_coeff(int n, int j) {
    int q = n >> 1, c = n & 1;
    int s = j >> 1, d = j & 1;
    int qs = q & s;
    float h = (((qs ^ (qs >> 1)) & 1) != 0) ? -1.0f : 1.0f;
    if ((q & 1) == 0) return (c == d) ? h : 0.0f;
    return (c != d) ? (c ? h : -h) : 0.0f;
}

#define WAVE_STRIDE 16384            // 2048 blocks * 8 waves

__global__ __launch_bounds__(256)
void hermit4_wmma_kernel(const float* __restrict__ x0,
                         const float* __restrict__ x1,
                         const float* __restrict__ x2,
                         const float* __restrict__ x3,
                         float* __restrict__ out,
                         int batch, int tiles, int tiles_per_wave)
{
    const int lane   = threadIdx.x & 31;
    const int waveId = (blockIdx.x * (blockDim.x >> 5)) + (threadIdx.x >> 5);
    const int m      = lane & 15;    // A row (pair-in-tile) / B column n
    const bool hi    = lane >= 16;   // lane half selects K pair 2,3 vs 0,1

    // --- constant B operands (per-lane, computed once) ----------------------
    // B 4x16 striping (consistent with A / the 8b & 16b B tables):
    //   VGPR0 (.x): K = hi?2:0 ; VGPR1 (.y): K = hi?3:1 ; column n = lane&15
    // B1 holds M8^T columns j = 0..3, B2 holds j = 4..7; columns n>=8 are 0.
    const int klo = hi ? 2 : 0;
    v2f b1 = {0.0f, 0.0f}, b2 = {0.0f, 0.0f};
    if (m < 8) {
        b1.x = m8_coeff(m, klo);
        b1.y = m8_coeff(m, klo + 1);
        b2.x = m8_coeff(m, klo + 4);
        b2.y = m8_coeff(m, klo + 5);
    }

    // --- hoisted per-lane load pointers (t = 0), uniform stride per tile ----
    // A VGPR0 = Re (K0 / K2), VGPR1 = Im (K1 / K3) == (v2f).x/.y of one b64.
    const float* a1ptr = (hi ? x1 : x0) + (size_t)(waveId * 16 + m) * 2;
    const float* a2ptr = (hi ? x3 : x2) + (size_t)(waveId * 16 + m) * 2;
    const size_t   lstep = (size_t)WAVE_STRIDE * 32;   // floats per tile step

    // --- hoisted 32-bit store offsets from the uniform `out` base -----------
    // Pair p's 8 output floats sit at out[(p&3)*batch*8 + (p>>2)*8 + n].
    // Advancing t adds WAVE_STRIDE*16 to p (l unchanged, b += WAVE_STRIDE*4),
    // i.e. a uniform +WAVE_STRIDE*32 to every offset.
    unsigned soff[8];
    {
        const int pbase0 = waveId * 16 + (hi ? 8 : 0);
#pragma unroll
        for (int r = 0; r < 8; ++r) {
            const int p = pbase0 + r;
            soff[r] = (unsigned)(((p & 3) * batch + (p >> 2)) * 8 + m);
        }
    }
    const unsigned sstep = (unsigned)WAVE_STRIDE * 32u;

    for (int t = 0; t < tiles_per_wave; ++t) {
        const int tile = waveId + t * WAVE_STRIDE;
        if (tile >= tiles) break;            // wave-uniform: EXEC stays all-1

        // ---- loads: one b64 per lane, 128B contiguous per half-wave --------
        v2f a1 = *(const v2f*)a1ptr;
        v2f a2 = *(const v2f*)a2ptr;

        // ---- D = A1*B1 + A2*B2 : exact f32 butterfly for 16 pairs ----------
        v8f d = {};
        d = __builtin_amdgcn_wmma_f32_16x16x4_f32(
                false, a1, false, b1, (short)0, d, false, false);
        d = __builtin_amdgcn_wmma_f32_16x16x4_f32(
                false, a2, false, b2, (short)0, d, false, false);

        // ---- stores: D VGPR r -> pair pbase+r, float n = m (n<8 valid) -----
        if (m < 8) {
#pragma unroll
            for (int r = 0; r < 8; ++r) {
                out[soff[r]] = d[r];
            }
        }

        // ---- uniform advance to next tile ----------------------------------
        a1ptr += lstep;
        a2ptr += lstep;
#pragma unroll
        for (int r = 0; r < 8; ++r) soff[r] += sstep;
    }
}

extern "C" void kernel_launch(void* const* d_in, const int* in_sizes, int n_in,
                              void* d_out, int out_size, void* d_ws, size_t ws_size,
                              hipStream_t stream) {
    (void)n_in; (void)out_size; (void)d_ws; (void)ws_size;
    const float* x0 = (const float*)d_in[0];
    const float* x1 = (const float*)d_in[1];
    const float* x2 = (const float*)d_in[2];
    const float* x3 = (const float*)d_in[3];
    float* out = (float*)d_out;

    const int batch = in_sizes[0] / 8;        // [B, 4, 2] f32
    const int pairs = batch * 4;              // (b, l) pairs
    const int tiles = (pairs + 15) / 16;      // 16 pairs per wave-tile

    const int blocks = 2048;                  // 8 waves/block -> 16384 waves
    const int waves  = blocks * 8;
    const int tiles_per_wave = (tiles + waves - 1) / waves;

    hipLaunchKernelGGL(hermit4_wmma_kernel, dim3(blocks), dim3(256), 0, stream,
                       x0, x1, x2, x3, out, batch, tiles, tiles_per_wave);
}